// LatentRNNAgent_6957847020143
// MI455X (gfx1250) — compile-verified
//
#include <hip/hip_runtime.h>
#include <hip/hip_bf16.h>
#include <math.h>

// ---------------------------------------------------------------------------
// LatentRNNAgent forward for MI455X (gfx1250), fp32 via V_WMMA_F32_16X16X4_F32
// All hot-loop addressing uses 32-bit element offsets from uniform base
// pointers so the backend can emit global_load saddr (SGPR base + VGPR off).
// ---------------------------------------------------------------------------

typedef __attribute__((ext_vector_type(2))) float v2f;
typedef __attribute__((ext_vector_type(8))) float v8f;

#define NROWS   4096      // BS * N_AGENTS
#define DLAT    64
#define HID     512
#define INP     256
#define NACT    64
#define NAGENTS 8

__device__ __forceinline__ v8f wmma_f32(v2f a, v2f b, v8f c) {
  // (neg_a, A, neg_b, B, c_mod, C, reuse_a, reuse_b)
  return __builtin_amdgcn_wmma_f32_16x16x4_f32(false, a, false, b, (short)0, c,
                                               false, false);
}

// ---------------------------------------------------------------------------
// Generic fp32 WMMA GEMM:  C[M x N] = act( A[M x K] @ B[K x N] + bias )
// A is optionally a concat of A1 (cols < splitK) and A2 (cols >= splitK);
// splitK must be a multiple of 4 so lane-halves never straddle the split.
// Tile: 64x64 per block, 8 waves (256 thr), 2 WMMA output tiles per wave.
// B is row-major (K x N), leading dim ldb.
// ---------------------------------------------------------------------------
__global__ __launch_bounds__(256) void gemm_wmma_f32(
    const float* __restrict__ A1, const float* __restrict__ A2, int splitK,
    int lda1, int lda2,
    const float* __restrict__ B, int ldb,
    const float* __restrict__ bias,
    float* __restrict__ C, int ldc,
    int K, int relu)
{
  const int lane = threadIdx.x & 31;
  const int wave = threadIdx.x >> 5;
  const int half = lane >> 4;       // K-pair select (A/B frags)
  const int l    = lane & 15;       // row (A) / col (B,C) within tile
  const int mi   = wave & 3;        // M sub-tile 0..3
  const int nb   = (wave >> 2) * 2; // N sub-tile base 0 or 2

  const unsigned row  = blockIdx.x * 64u + mi * 16 + l;
  const unsigned col0 = blockIdx.y * 64u + nb * 16 + l;
  const unsigned col1 = col0 + 16;
  const unsigned uldb = (unsigned)ldb;

  v8f acc0 = {}; v8f acc1 = {};

  if (splitK >= K) {
    // Fast path: single A source, pure 32-bit offsets -> saddr loads.
    const unsigned abase = row * (unsigned)lda1;
    for (int k = 0; k < K; k += 4) {
      const unsigned k0 = (unsigned)k + half * 2;
      v2f a; a.x = A1[abase + k0]; a.y = A1[abase + k0 + 1];
      const unsigned boff = k0 * uldb;
      v2f b0; b0.x = B[boff + col0]; b0.y = B[boff + uldb + col0];
      v2f b1; b1.x = B[boff + col1]; b1.y = B[boff + uldb + col1];
      acc0 = wmma_f32(a, b0, acc0);
      acc1 = wmma_f32(a, b1, acc1);
    }
  } else {
    // Split-A path (inf1 concat of [h_in | x_in[:, :INP-8]]).
    const unsigned abase1 = row * (unsigned)lda1;
    const unsigned abase2 = row * (unsigned)lda2;
    for (int k = 0; k < K; k += 4) {
      const unsigned k0 = (unsigned)k + half * 2;
      v2f a;
      if ((int)k0 < splitK) {
        a.x = A1[abase1 + k0];
        a.y = A1[abase1 + k0 + 1];
      } else {
        a.x = A2[abase2 + (k0 - splitK)];
        a.y = A2[abase2 + (k0 - splitK) + 1];
      }
      const unsigned boff = k0 * uldb;
      v2f b0; b0.x = B[boff + col0]; b0.y = B[boff + uldb + col0];
      v2f b1; b1.x = B[boff + col1]; b1.y = B[boff + uldb + col1];
      acc0 = wmma_f32(a, b0, acc0);
      acc1 = wmma_f32(a, b1, acc1);
    }
  }

  // C/D layout: VGPR i, lane-half h -> row = i + 8*h, col = lane&15
  const float bv0 = bias ? bias[col0] : 0.0f;
  const float bv1 = bias ? bias[col1] : 0.0f;
#pragma unroll
  for (int i = 0; i < 8; ++i) {
    const unsigned r = blockIdx.x * 64u + mi * 16 + half * 8 + i;
    float c0 = acc0[i] + bv0;
    float c1 = acc1[i] + bv1;
    if (relu) { c0 = fmaxf(c0, 0.0f); c1 = fmaxf(c1, 0.0f); }
    C[r * (unsigned)ldc + col0] = c0;
    C[r * (unsigned)ldc + col1] = c1;
  }
}

// ---------------------------------------------------------------------------
// Tiny embed: emb[8 x 128] = x_in[:8, -8:] @ embed_w[8 x 128] + embed_b
// ---------------------------------------------------------------------------
__global__ __launch_bounds__(1024) void embed_kernel(
    const float* __restrict__ x_in, const float* __restrict__ embed_w,
    const float* __restrict__ embed_b,
    float* __restrict__ mu_e, float* __restrict__ logv_e)
{
  const int idx = threadIdx.x;       // 0..1023
  const int a = idx >> 7;            // agent 0..7
  const int j = idx & 127;           // output col 0..127
  float s = embed_b[j];
#pragma unroll
  for (int t = 0; t < NAGENTS; ++t)
    s += x_in[a * INP + (INP - NAGENTS) + t] * embed_w[t * 128 + j];
  if (j < DLAT) mu_e[a * DLAT + j] = s;
  else          logv_e[a * DLAT + (j - DLAT)] = s;
}

// ---------------------------------------------------------------------------
// latent + per-row loss contribution.
// entropy+kl per row simplifies to 32*ln(2pi) + sum_d 0.5*(ve/vi + dmu^2/vi + lvi)
// ---------------------------------------------------------------------------
__global__ __launch_bounds__(64) void latent_loss_kernel(
    const float* __restrict__ eps, const float* __restrict__ inf2o,
    const float* __restrict__ mu_e_t, const float* __restrict__ logv_e_t,
    float* __restrict__ latent, float* __restrict__ lossbuf)
{
  const unsigned n = blockIdx.x;
  const int d = threadIdx.x;          // 0..63
  const int a = n & (NAGENTS - 1);

  const float mu_e = mu_e_t[a * DLAT + d];
  const float lve  = logv_e_t[a * DLAT + d];
  const float ve   = expf(lve);
  latent[n * DLAT + d] = mu_e + sqrtf(ve) * eps[n * DLAT + d];

  const float mu_i = inf2o[n * 128 + d];
  const float lvi  = inf2o[n * 128 + DLAT + d];
  const float vi   = expf(lvi);
  const float dm   = mu_i - mu_e;
  float c = 0.5f * (ve / vi + dm * dm / vi + lvi);

  for (int off = 16; off > 0; off >>= 1) c += __shfl_down(c, off, 32);
  __shared__ float sred[2];
  if ((threadIdx.x & 31) == 0) sred[threadIdx.x >> 5] = c;
  __syncthreads();
  if (threadIdx.x == 0) lossbuf[n] = sred[0] + sred[1];
}

__global__ __launch_bounds__(256) void loss_reduce_kernel(
    const float* __restrict__ lossbuf, float* __restrict__ loss_out)
{
  float s = 0.0f;
  for (int i = threadIdx.x; i < NROWS; i += 256) s += lossbuf[i];
  for (int off = 16; off > 0; off >>= 1) s += __shfl_down(s, off, 32);
  __shared__ float sm[8];
  if ((threadIdx.x & 31) == 0) sm[threadIdx.x >> 5] = s;
  __syncthreads();
  if (threadIdx.x == 0) {
    float t = 0.0f;
#pragma unroll
    for (int i = 0; i < 8; ++i) t += sm[i];
    // per-row constant: 0.5*D*(1+ln 2pi) - 0.5*D = 32*ln(2pi), times 4096 rows
    t += (float)NROWS * 32.0f * logf(2.0f * 3.14159265358979323846f);
    *loss_out = t;
  }
}

// ---------------------------------------------------------------------------
// GRU gate elementwise
// ---------------------------------------------------------------------------
__global__ __launch_bounds__(256) void gru_kernel(
    const float* __restrict__ gi, const float* __restrict__ gh,
    const float* __restrict__ h_in, float* __restrict__ h_out)
{
  const unsigned idx = blockIdx.x * 256u + threadIdx.x; // < 4096*512
  const unsigned n = idx >> 9;
  const unsigned j = idx & 511;
  const unsigned g = n * (3 * HID) + j;
  const float r  = 1.0f / (1.0f + expf(-(gi[g] + gh[g])));
  const float z  = 1.0f / (1.0f + expf(-(gi[g + HID] + gh[g + HID])));
  const float nn = tanhf(gi[g + 2 * HID] + r * gh[g + 2 * HID]);
  h_out[idx] = (1.0f - z) * nn + z * h_in[idx];
}

// ---------------------------------------------------------------------------
// Hypernetwork q partial:
//   qpart[kc][n,a] = sum_{k in chunk} lat[n,k] * (sum_h h[n,h]*Wnn[k,h,a])
// 64-row H tile in LDS (padded stride 516 to break 512-stride bank conflicts),
// inner GEMM (64x512)@(512x64) per k via f32 WMMA. Grid: 64 n-tiles x 8 chunks.
// ---------------------------------------------------------------------------
#define QM   64
#define QHP  516                         // padded H row stride (floats)
#define QLDS (QM * QHP * 4)              // 132096 bytes

__global__ __launch_bounds__(512) void q_partial_kernel(
    const float* __restrict__ h, const float* __restrict__ lat,
    const float* __restrict__ Wnn, float* __restrict__ qpart)
{
  extern __shared__ float hs[];          // [QM][QHP]
  const int tid = threadIdx.x;
  const unsigned block_row = blockIdx.x * QM;
  const unsigned kchunk = blockIdx.y;    // 0..7

  // stage H tile: 64x512 floats via float4
  for (int idx = tid; idx < QM * (HID / 4); idx += 512) {
    const int r  = idx >> 7;             // HID/4 = 128 float4 per row
    const int c4 = (idx & 127) << 2;
    const float4 v = *(const float4*)&h[(block_row + r) * HID + c4];
    *(float4*)&hs[r * QHP + c4] = v;
  }
  __syncthreads();

  const int lane = tid & 31, wave = tid >> 5;
  const int half = lane >> 4, l = lane & 15;
  const int mi = wave >> 2;              // 0..3 (M sub-tile)
  const int ai = wave & 3;               // 0..3 (action sub-tile)

  const float* hrow = &hs[(mi * 16 + l) * QHP];
  const unsigned acol = ai * 16 + l;

  v8f qacc = {};
  const unsigned k0 = kchunk * 32;
  // lat fold base: row = block_row + mi*16 + i + 8*half, col = k
  const unsigned latbase = (block_row + mi * 16 + half * 8) * 256u;

  for (unsigned k = k0; k < k0 + 32; ++k) {
    const unsigned wbase = k * (unsigned)(HID * NACT) + acol;  // Wnn[k][h][a]
    v8f t = {};
    for (int kk = 0; kk < HID; kk += 4) {
      const unsigned kA = (unsigned)kk + half * 2;
      v2f a = *(const v2f*)&hrow[kA];                 // ds_load_b64
      v2f b; b.x = Wnn[wbase + kA * NACT];
             b.y = Wnn[wbase + (kA + 1) * NACT];
      t = wmma_f32(a, b, t);
    }
#pragma unroll
    for (int i = 0; i < 8; ++i) qacc[i] += lat[latbase + i * 256u + k] * t[i];
  }

  const unsigned obase = kchunk * (unsigned)(NROWS * NACT);
#pragma unroll
  for (int i = 0; i < 8; ++i) {
    const unsigned n = block_row + mi * 16 + half * 8 + i;
    qpart[obase + n * NACT + acol] = qacc[i];
  }
}

__global__ __launch_bounds__(256) void q_final_kernel(
    const float* __restrict__ qpart, const float* __restrict__ fc2b,
    const float* __restrict__ hb, float* __restrict__ q_out)
{
  const unsigned idx = blockIdx.x * 256u + threadIdx.x; // < 4096*64
  float s = fc2b[idx] + hb[idx];
#pragma unroll
  for (int c = 0; c < 8; ++c) s += qpart[c * (unsigned)(NROWS * NACT) + idx];
  q_out[idx] = s;
}

// ---------------------------------------------------------------------------
// Host launcher
// ---------------------------------------------------------------------------
extern "C" void kernel_launch(void* const* d_in, const int* in_sizes, int n_in,
                              void* d_out, int out_size, void* d_ws, size_t ws_size,
                              hipStream_t stream) {
  const float* x_in    = (const float*)d_in[0];
  const float* h_in    = (const float*)d_in[1];
  const float* eps     = (const float*)d_in[2];
  // d_in[3] = bs (scalar, fixed 512)
  const float* embed_w = (const float*)d_in[4];
  const float* embed_b = (const float*)d_in[5];
  const float* inf1_w  = (const float*)d_in[6];
  const float* inf1_b  = (const float*)d_in[7];
  const float* inf2_w  = (const float*)d_in[8];
  const float* inf2_b  = (const float*)d_in[9];
  const float* lat1_w  = (const float*)d_in[10];
  const float* lat1_b  = (const float*)d_in[11];
  const float* lat2_w  = (const float*)d_in[12];
  const float* lat2_b  = (const float*)d_in[13];
  const float* fc1_w   = (const float*)d_in[14];
  const float* fc1_b   = (const float*)d_in[15];
  const float* gru_wih = (const float*)d_in[16];
  const float* gru_bih = (const float*)d_in[17];
  const float* gru_whh = (const float*)d_in[18];
  const float* gru_bhh = (const float*)d_in[19];
  const float* w_nn_w  = (const float*)d_in[20];
  const float* w_nn_b  = (const float*)d_in[21];
  const float* b_nn_w  = (const float*)d_in[22];
  const float* b_nn_b  = (const float*)d_in[23];

  float* q_out    = (float*)d_out;
  float* h_out    = q_out + (size_t)NROWS * NACT;
  float* loss_out = h_out + (size_t)NROWS * HID;

  float* ws        = (float*)d_ws;
  float* ws_mu     = ws;                                   // 8*64
  float* ws_logv   = ws_mu     + NAGENTS * DLAT;           // 8*64
  float* ws_inf1   = ws_logv   + NAGENTS * DLAT;           // 4096*256
  float* ws_inf2   = ws_inf1   + (size_t)NROWS * 256;      // 4096*128
  float* ws_latent = ws_inf2   + (size_t)NROWS * 128;      // 4096*64
  float* ws_lata   = ws_latent + (size_t)NROWS * DLAT;     // 4096*256
  float* ws_lat    = ws_lata   + (size_t)NROWS * 256;      // 4096*256
  float* ws_x1     = ws_lat    + (size_t)NROWS * 256;      // 4096*512
  float* ws_gi     = ws_x1     + (size_t)NROWS * HID;      // 4096*1536
  float* ws_gh     = ws_gi     + (size_t)NROWS * 3 * HID;  // 4096*1536
  float* ws_fc2b   = ws_gh     + (size_t)NROWS * 3 * HID;  // 4096*64
  float* ws_hb     = ws_fc2b   + (size_t)NROWS * NACT;     // 4096*64
  float* ws_qpart  = ws_hb     + (size_t)NROWS * NACT;     // 8*4096*64
  float* ws_lossb  = ws_qpart  + (size_t)8 * NROWS * NACT; // 4096

  const int BIG = 1 << 30;   // splitK sentinel: always read A1
  dim3 blk(256);

  // --- embed (tiny) ---
  embed_kernel<<<1, 1024, 0, stream>>>(x_in, embed_w, embed_b, ws_mu, ws_logv);

  // --- fc1: x1 = relu(x @ fc1_w + b), (4096x256)@(256x512) ---
  gemm_wmma_f32<<<dim3(NROWS/64, HID/64), blk, 0, stream>>>(
      x_in, x_in, BIG, INP, INP, fc1_w, HID, fc1_b, ws_x1, HID, INP, 1);

  // --- GRU input/hidden GEMMs: (4096x512)@(512x1536) ---
  gemm_wmma_f32<<<dim3(NROWS/64, (3*HID)/64), blk, 0, stream>>>(
      ws_x1, ws_x1, BIG, HID, HID, gru_wih, 3*HID, gru_bih, ws_gi, 3*HID, HID, 0);
  gemm_wmma_f32<<<dim3(NROWS/64, (3*HID)/64), blk, 0, stream>>>(
      h_in, h_in, BIG, HID, HID, gru_whh, 3*HID, gru_bhh, ws_gh, 3*HID, HID, 0);

  // --- GRU gates -> h_out ---
  gru_kernel<<<(NROWS * HID) / 256, blk, 0, stream>>>(ws_gi, ws_gh, h_in, h_out);

  // --- inference net: inf1 fuses concat([h_in, x_in[:, :248]]), K=760 ---
  gemm_wmma_f32<<<dim3(NROWS/64, 256/64), blk, 0, stream>>>(
      h_in, x_in, HID, HID, INP, inf1_w, 256, inf1_b, ws_inf1, 256, HID + INP - NAGENTS, 1);
  gemm_wmma_f32<<<dim3(NROWS/64, 128/64), blk, 0, stream>>>(
      ws_inf1, ws_inf1, BIG, 256, 256, inf2_w, 128, inf2_b, ws_inf2, 128, 256, 0);

  // --- latent sample + loss ---
  latent_loss_kernel<<<NROWS, 64, 0, stream>>>(eps, ws_inf2, ws_mu, ws_logv,
                                               ws_latent, ws_lossb);
  loss_reduce_kernel<<<1, 256, 0, stream>>>(ws_lossb, loss_out);

  // --- latent MLP ---
  gemm_wmma_f32<<<dim3(NROWS/64, 256/64), blk, 0, stream>>>(
      ws_latent, ws_latent, BIG, DLAT, DLAT, lat1_w, 256, lat1_b, ws_lata, 256, DLAT, 1);
  gemm_wmma_f32<<<dim3(NROWS/64, 256/64), blk, 0, stream>>>(
      ws_lata, ws_lata, BIG, 256, 256, lat2_w, 256, lat2_b, ws_lat, 256, 256, 1);

  // --- fc2 bias path: fc2b = lat @ b_nn_w + b_nn_b ---
  gemm_wmma_f32<<<dim3(NROWS/64, NACT/64), blk, 0, stream>>>(
      ws_lat, ws_lat, BIG, 256, 256, b_nn_w, NACT, b_nn_b, ws_fc2b, NACT, 256, 0);

  // --- w_nn bias contraction: hb = h @ w_nn_b(512x64) ---
  gemm_wmma_f32<<<dim3(NROWS/64, NACT/64), blk, 0, stream>>>(
      h_out, h_out, BIG, HID, HID, w_nn_b, NACT, nullptr, ws_hb, NACT, HID, 0);

  // --- hypernetwork q (dominant 68.7 GFLOP), 132KB LDS per block ---
  hipFuncSetAttribute(reinterpret_cast<const void*>(&q_partial_kernel),
                      hipFuncAttributeMaxDynamicSharedMemorySize, QLDS);
  q_partial_kernel<<<dim3(NROWS/QM, 8), dim3(512), QLDS, stream>>>(
      h_out, ws_lat, w_nn_w, ws_qpart);
  q_final_kernel<<<(NROWS * NACT) / 256, blk, 0, stream>>>(
      ws_qpart, ws_fc2b, ws_hb, q_out);
}